// BahdanauAttention_87325275062388
// MI455X (gfx1250) — compile-verified
//
#include <hip/hip_runtime.h>
#include <math.h>

// Problem dims (fixed by reference)
#define B   32
#define TK  4096
#define H   1024
#define DV  1024
#define QS  1024
#define EPS 1e-12f

#define TCH 8              // t-chunks for context partial sums
#define TCHUNK (TK / TCH)  // 512

typedef __attribute__((ext_vector_type(2))) float v2f;
typedef __attribute__((ext_vector_type(8))) float v8f;

// Hardware tanh (V_TANH_F32, single TRANS op on gfx1250) with libm fallback.
#if defined(__has_builtin)
#  if __has_builtin(__builtin_amdgcn_tanhf)
#    define FAST_TANH(x) __builtin_amdgcn_tanhf(x)
#  else
#    define FAST_TANH(x) tanhf(x)
#  endif
#else
#  define FAST_TANH(x) tanhf(x)
#endif

// ---------------------------------------------------------------------------
// Kernel A: q[b,h] = sum_s query[b,s] * Wq[h,s]   (GEMM M=32,N=1024,K=1024)
// One wave per 16x16 C tile, K stepped by 4 via V_WMMA_F32_16X16X4_F32.
// A (16x4 f32): lanes 0-15 hold M=0..15; VGPR0 = K=0 (lo lanes) / K=2 (hi),
// VGPR1 = K=1 / K=3.  B (4x16 f32): VGPR0 = row K=0 / K=2, VGPR1 = K=1 / K=3,
// N striped across lanes.  C (16x16 f32): VGPR i -> M=i (lo lanes) / M=i+8.
// ---------------------------------------------------------------------------
__global__ void qproj_wmma(const float* __restrict__ query,  // [B,QS]
                           const float* __restrict__ Wq,     // [H,QS]
                           float* __restrict__ qout) {       // [B,H]
    const int lane     = threadIdx.x;        // 0..31, exactly one wave
    const int laneHalf = lane & 15;
    const int kOff     = (lane >> 4) << 1;   // 0 for lanes 0-15, 2 for 16-31
    const int nBase    = blockIdx.x * 16;    // tile along H
    const int mBase    = blockIdx.y * 16;    // tile along B

    const float* qrow = query + (size_t)(mBase + laneHalf) * QS + kOff;
    const float* wrow = Wq    + (size_t)(nBase + laneHalf) * QS + kOff;

    v8f acc = {};
    #pragma unroll 4
    for (int k = 0; k < QS; k += 4) {
        v2f a, b;
        a[0] = qrow[k];     a[1] = qrow[k + 1];   // A[m, k+kOff .. k+kOff+1]
        b[0] = wrow[k];     b[1] = wrow[k + 1];   // B[k+kOff.., n] = Wq[n, s]
        acc = __builtin_amdgcn_wmma_f32_16x16x4_f32(
            /*neg_a=*/false, a, /*neg_b=*/false, b,
            /*c_mod=*/(short)0, acc, /*reuse_a=*/false, /*reuse_b=*/false);
    }

    const int n    = nBase + laneHalf;
    const int mAdd = (lane >> 4) * 8;
    #pragma unroll
    for (int i = 0; i < 8; ++i)
        qout[(size_t)(mBase + mAdd + i) * H + n] = acc[i];
}

// ---------------------------------------------------------------------------
// Kernel B: scores[b,t] = sum_h v_energy[h] * tanh(q[b,h] + proj_key[b,t,h])
// 8 waves per block, one wave per t-row; streams 512 MB of proj_key.
// Hardware v_tanh_f32 keeps this pass under the 23.3 TB/s memory roof.
// ---------------------------------------------------------------------------
__global__ void scores_kernel(const float* __restrict__ proj_key, // [B,TK,H]
                              const float* __restrict__ qproj,    // [B,H]
                              const float* __restrict__ v_energy, // [H]
                              float* __restrict__ scores) {       // [B,TK]
    __shared__ float s_q[H];
    __shared__ float s_v[H];

    const int rowsPerBlock = 8;
    const int b  = blockIdx.x / (TK / rowsPerBlock);
    const int t0 = (blockIdx.x % (TK / rowsPerBlock)) * rowsPerBlock;

    for (int i = threadIdx.x; i < H; i += blockDim.x) {
        s_q[i] = qproj[(size_t)b * H + i];
        s_v[i] = v_energy[i];
    }
    __syncthreads();

    const int wave = threadIdx.x >> 5;
    const int lane = threadIdx.x & 31;
    const int t    = t0 + wave;

    const float4* pk4 = (const float4*)(proj_key + ((size_t)b * TK + t) * H);
    const float4* q4  = (const float4*)s_q;
    const float4* v4  = (const float4*)s_v;

    // Speculative prefetch of this wave's row 8 t-steps ahead
    // (global_prefetch_b8; OOB speculative prefetches are silently dropped).
    if (t + rowsPerBlock < TK)
        __builtin_prefetch(pk4 + (size_t)rowsPerBlock * (H / 4) + lane, 0, 0);

    float acc = 0.f;
    #pragma unroll
    for (int c = 0; c < H / 128; ++c) {       // 8 coalesced float4 passes
        const int idx = c * 32 + lane;
        float4 p  = pk4[idx];
        float4 qq = q4[idx];
        float4 vv = v4[idx];
        acc += vv.x * FAST_TANH(qq.x + p.x);
        acc += vv.y * FAST_TANH(qq.y + p.y);
        acc += vv.z * FAST_TANH(qq.z + p.z);
        acc += vv.w * FAST_TANH(qq.w + p.w);
    }
    #pragma unroll
    for (int off = 16; off > 0; off >>= 1)
        acc += __shfl_down(acc, off, 32);
    if (lane == 0) scores[(size_t)b * TK + t] = acc;
}

// ---------------------------------------------------------------------------
// Kernel C: masked softmax -> +copy / L1 renorm -> +ratings / re-softmax.
// One 1024-thread block per batch, 4 elements per thread, LDS reductions.
// ---------------------------------------------------------------------------
__global__ void softmax_chain(const float* __restrict__ scores,   // [B,TK]
                              const int*   __restrict__ mask,     // [B,TK]
                              const float* __restrict__ copyv,    // [B,TK]
                              const float* __restrict__ ratings,  // [B,TK]
                              float* __restrict__ alphas_out) {   // [B,TK]
    const int b   = blockIdx.x;
    const int tid = threadIdx.x;   // 0..1023
    __shared__ float red[1024];

    float x[4];
    #pragma unroll
    for (int i = 0; i < 4; ++i) {
        const size_t t = (size_t)b * TK + tid + i * 1024;
        x[i] = (mask[t] == 0) ? -INFINITY : scores[t];
    }

    // --- softmax #1: max ---
    float m = fmaxf(fmaxf(x[0], x[1]), fmaxf(x[2], x[3]));
    red[tid] = m; __syncthreads();
    for (int s = 512; s >= 1; s >>= 1) {
        if (tid < s) red[tid] = fmaxf(red[tid], red[tid + s]);
        __syncthreads();
    }
    m = red[0]; __syncthreads();

    // --- softmax #1: exp + sum ---
    float lsum = 0.f;
    #pragma unroll
    for (int i = 0; i < 4; ++i) { x[i] = expf(x[i] - m); lsum += x[i]; }
    red[tid] = lsum; __syncthreads();
    for (int s = 512; s >= 1; s >>= 1) {
        if (tid < s) red[tid] += red[tid + s];
        __syncthreads();
    }
    const float inv1 = 1.f / red[0]; __syncthreads();

    // --- + copy, L1 normalize ---
    float l1p = 0.f;
    #pragma unroll
    for (int i = 0; i < 4; ++i) {
        const size_t t = (size_t)b * TK + tid + i * 1024;
        x[i] = x[i] * inv1 + copyv[t];
        l1p += fabsf(x[i]);
    }
    red[tid] = l1p; __syncthreads();
    for (int s = 512; s >= 1; s >>= 1) {
        if (tid < s) red[tid] += red[tid + s];
        __syncthreads();
    }
    const float invL1 = 1.f / fmaxf(red[0], EPS); __syncthreads();

    // --- + ratings, softmax #2 ---
    float m2p = -INFINITY;
    #pragma unroll
    for (int i = 0; i < 4; ++i) {
        const size_t t = (size_t)b * TK + tid + i * 1024;
        x[i] = x[i] * invL1 + ratings[t];
        m2p = fmaxf(m2p, x[i]);
    }
    red[tid] = m2p; __syncthreads();
    for (int s = 512; s >= 1; s >>= 1) {
        if (tid < s) red[tid] = fmaxf(red[tid], red[tid + s]);
        __syncthreads();
    }
    const float m2 = red[0]; __syncthreads();

    float lsum2 = 0.f;
    #pragma unroll
    for (int i = 0; i < 4; ++i) { x[i] = expf(x[i] - m2); lsum2 += x[i]; }
    red[tid] = lsum2; __syncthreads();
    for (int s = 512; s >= 1; s >>= 1) {
        if (tid < s) red[tid] += red[tid + s];
        __syncthreads();
    }
    const float inv2 = 1.f / red[0];

    #pragma unroll
    for (int i = 0; i < 4; ++i) {
        const size_t t = (size_t)b * TK + tid + i * 1024;
        alphas_out[t] = x[i] * inv2;
    }
}

// ---------------------------------------------------------------------------
// Kernel D: partial context over a t-chunk: streams 512 MB of value.
// grid (B, TCH), block 256; each thread owns a float4 of DV.
// ---------------------------------------------------------------------------
__global__ void context_partial(const float* __restrict__ value,   // [B,TK,DV]
                                const float* __restrict__ alphas,  // [B,TK]
                                float* __restrict__ partial) {     // [B,TCH,DV]
    const int b   = blockIdx.x;
    const int tc  = blockIdx.y;
    const int tid = threadIdx.x;   // 0..255
    __shared__ float s_a[TCHUNK];

    for (int i = tid; i < TCHUNK; i += 256)
        s_a[i] = alphas[(size_t)b * TK + (size_t)tc * TCHUNK + i];
    __syncthreads();

    const float4* v4 =
        (const float4*)(value + ((size_t)b * TK + (size_t)tc * TCHUNK) * DV);
    float4 acc = make_float4(0.f, 0.f, 0.f, 0.f);
    for (int tt = 0; tt < TCHUNK; ++tt) {
        const float a = s_a[tt];
        const float4 v = v4[(size_t)tt * (DV / 4) + tid];
        acc.x += a * v.x; acc.y += a * v.y;
        acc.z += a * v.z; acc.w += a * v.w;
    }
    float4* p4 = (float4*)(partial + (size_t)(b * TCH + tc) * DV);
    p4[tid] = acc;
}

// Kernel E: reduce partials -> context (deterministic, no float atomics)
__global__ void context_reduce(const float* __restrict__ partial, // [B,TCH,DV]
                               float* __restrict__ ctx) {         // [B,DV]
    const int idx = blockIdx.x * blockDim.x + threadIdx.x; // 0..B*DV-1
    const int b = idx / DV, d = idx % DV;
    float s = 0.f;
    #pragma unroll
    for (int c = 0; c < TCH; ++c)
        s += partial[(size_t)(b * TCH + c) * DV + d];
    ctx[idx] = s;
}

// ---------------------------------------------------------------------------
extern "C" void kernel_launch(void* const* d_in, const int* in_sizes, int n_in,
                              void* d_out, int out_size, void* d_ws, size_t ws_size,
                              hipStream_t stream) {
    const float* query    = (const float*)d_in[0];   // [B,1,QS]
    const float* proj_key = (const float*)d_in[1];   // [B,TK,H]
    const float* value    = (const float*)d_in[2];   // [B,TK,DV]
    const int*   mask     = (const int*)  d_in[3];   // [B,TK]
    const float* copyv    = (const float*)d_in[4];   // [B,TK]
    const float* ratings  = (const float*)d_in[5];   // [B,TK]
    const float* Wq       = (const float*)d_in[6];   // [H,QS]
    const float* v_energy = (const float*)d_in[7];   // [H]

    float* out    = (float*)d_out;
    float* ctx    = out;             // context [B,1,DV] first in return order
    float* alphas = out + B * DV;    // alphas  [B,1,TK]

    float* ws      = (float*)d_ws;
    float* qproj   = ws;                 // B*H      = 32768 floats
    float* scores  = qproj + B * H;      // B*TK     = 131072 floats
    float* partial = scores + B * TK;    // B*TCH*DV = 262144 floats

    qproj_wmma     <<<dim3(H / 16, B / 16), 32,   0, stream>>>(query, Wq, qproj);
    scores_kernel  <<<dim3(B * TK / 8),     256,  0, stream>>>(proj_key, qproj,
                                                               v_energy, scores);
    softmax_chain  <<<dim3(B),              1024, 0, stream>>>(scores, mask, copyv,
                                                               ratings, alphas);
    context_partial<<<dim3(B, TCH),         256,  0, stream>>>(value, alphas, partial);
    context_reduce <<<dim3(B * DV / 256),   256,  0, stream>>>(partial, ctx);
}